// LaterallyConnectedLayer3_75136157876878
// MI455X (gfx1250) — compile-verified
//
#include <hip/hip_runtime.h>

// LaterallyConnectedLayer3 on MI455X (gfx1250, wave32, WMMA).
// Pipeline: normK -> Ksum -> sumA -> conv1(WMMA f32, fused min/max/sum redux)
//           -> winner select -> conv2(WMMA f32, in-LDS kernel gather) -> out.
// Implicit GEMM with V_WMMA_F32_16X16X4_F32 (exact fp32; winner/minmax branch
// is numerically sensitive).  ~40 GFLOP vs ~100 MB HBM -> compute bound.

#define BATCH 64
#define NFM   32
#define CCH   64
#define TAPS  25
#define HW    4096

// padded LDS strides (even -> 8B alignment for v2f; stride%32==2 -> conflict-free)
#define CIP   34   // patch ci stride
#define CIW   18   // weight-slab ci stride

typedef float v2f __attribute__((ext_vector_type(2)));
typedef float v8f __attribute__((ext_vector_type(8)));

// order-preserving float <-> uint mapping for atomic min/max
__device__ __forceinline__ unsigned f2ord(float f) {
    unsigned u = __float_as_uint(f);
    return (u & 0x80000000u) ? ~u : (u | 0x80000000u);
}
__device__ __forceinline__ float ord2f(unsigned u) {
    return __uint_as_float((u & 0x80000000u) ? (u & 0x7FFFFFFFu) : ~u);
}

__global__ void k_init(unsigned* minU, unsigned* maxU, float* sumL) {
    int i = blockIdx.x * 256 + threadIdx.x;
    if (i < BATCH * CCH) { minU[i] = 0xFFFFFFFFu; maxU[i] = 0u; sumL[i] = 0.0f; }
}

// per (co,ci): minmax-normalize the 5x5 map.  Kn layout: [co][ci][tap]
__global__ void k_normK(const float* __restrict__ K, float* __restrict__ Kn) {
    int i = blockIdx.x * 256 + threadIdx.x;
    if (i >= CCH * CCH) return;
    const float* src = K + (size_t)i * TAPS;
    float mn = src[0], mx = src[0];
    for (int t = 1; t < TAPS; ++t) { float v = src[t]; mn = fminf(mn, v); mx = fmaxf(mx, v); }
    float inv = 1.0f / ((mx > mn) ? (mx - mn) : 1.0f);
    float* dst = Kn + (size_t)i * TAPS;
    for (int t = 0; t < TAPS; ++t) dst[t] = (src[t] - mn) * inv;
}

// fold replica pairs (A_rep = tile(A,2)): Ksum layout [co][tap][ci32]
__global__ void k_ksum(const float* __restrict__ Kn, float* __restrict__ Ksum) {
    int i = blockIdx.x * 256 + threadIdx.x;
    if (i >= CCH * TAPS * NFM) return;
    int ci = i & 31; int t = (i >> 5) % TAPS; int co = i / (TAPS * 32);
    Ksum[i] = Kn[(co * CCH + ci) * TAPS + t] + Kn[(co * CCH + ci + NFM) * TAPS + t];
}

// sumA[b,f] = sum over spatial of A
__global__ void k_sumA(const float* __restrict__ A, float* __restrict__ sumA) {
    int bf = blockIdx.x;
    const float* src = A + (size_t)bf * HW;
    float s = 0.0f;
    for (int i = threadIdx.x; i < HW; i += 256) s += src[i];
    for (int m = 16; m >= 1; m >>= 1) s += __shfl_xor(s, m, 32);
    __shared__ float part[8];
    if ((threadIdx.x & 31) == 0) part[threadIdx.x >> 5] = s;
    __syncthreads();
    if (threadIdx.x == 0) {
        float t = 0.0f;
        for (int w = 0; w < 8; ++w) t += part[w];
        sumA[bf] = t;
    }
}

// replica winner per (b, fm):  score = 0.9*sumA + 0.1*sum(minmax(L1*S))
__global__ void k_win(const unsigned* __restrict__ minU, const unsigned* __restrict__ maxU,
                      const float* __restrict__ sumL, const float* __restrict__ sumA,
                      int* __restrict__ win) {
    int i = blockIdx.x * 256 + threadIdx.x;
    if (i >= BATCH * NFM) return;
    int b = i / NFM, f = i % NFM;
    float sc[2];
    for (int r = 0; r < 2; ++r) {
        int c = b * CCH + r * NFM + f;
        float mn = ord2f(minU[c]), mx = ord2f(maxU[c]);
        float d = (mx > mn) ? (mx - mn) : 1.0f;
        float sL1n = (sumL[c] - (float)HW * mn) / d;
        sc[r] = 0.9f * sumA[i] + 0.1f * sL1n;
    }
    win[i] = (sc[1] > sc[0]) ? 1 : 0;  // argmax, first-max tie-break -> replica 0
}

// Implicit-GEMM conv, V_WMMA_F32_16X16X4_F32: M=16 out-ch, N=16 pixels, K=4 in-ch.
// Block = 256 threads = 8 waves; wave w handles output row y0+w, cols x0..x0+15,
// channels co0..co0+15.  MODE 0: conv1 (weights=Ksum, fused L1*S min/max/sum).
// MODE 1: conv2 (weights gathered from Kn via win while staging), writes d_out.
template <int MODE>
__global__ __launch_bounds__(256) void k_conv(
        const float* __restrict__ A, const float* __restrict__ Wt,
        const float* __restrict__ S, const int* __restrict__ win,
        float* __restrict__ out,
        unsigned* __restrict__ minU, unsigned* __restrict__ maxU,
        float* __restrict__ sumL) {
    // patch layout [rl(12)][cl(20)][ci pad 34] -> single ds_load_b64 B-frags
    __shared__ float patch[12 * 20 * CIP];         // 32640 B
    // weight slab layout [col(16)][tap(25)][ciHalf pad 18] -> ds_load_b64 A-frags
    __shared__ float wlds[16 * TAPS * CIW];        // 28800 B
    __shared__ float red[3][8][16];                //  1536 B
    __shared__ int   winsh[NFM];

    const int tid = threadIdx.x;
    const int wv  = tid >> 5;
    const int l   = tid & 31;
    const int lo  = l & 15;
    const int hi  = l >> 4;

    const int b   = blockIdx.z;
    const int co0 = blockIdx.y * 16;
    const int y0  = (blockIdx.x >> 2) * 8;
    const int x0  = (blockIdx.x & 3) * 16;

    if (MODE == 1 && tid < NFM) winsh[tid] = win[b * NFM + tid];

    // stage input patch: rows y0-2..y0+9, cols x0-2..x0+17, all 32 channels.
    // enumeration keeps global reads coalesced (cl inner); LDS stride CIP=34
    // makes the scattered stores only 2-way bank conflicted.
    const float* Ab = A + (size_t)b * NFM * HW;
    for (int j = tid; j < 12 * 32 * 20; j += 256) {
        int cl = j % 20; int ci = (j / 20) & 31; int rl = j / 640;
        int gy = y0 - 2 + rl, gx = x0 - 2 + cl;
        float v = 0.0f;
        if ((unsigned)gy < 64u && (unsigned)gx < 64u) v = Ab[(ci * 64 + gy) * 64 + gx];
        patch[(rl * 20 + cl) * CIP + ci] = v;
    }

    v8f acc = {0.f, 0.f, 0.f, 0.f, 0.f, 0.f, 0.f, 0.f};

    for (int half = 0; half < 2; ++half) {
        __syncthreads();
        // stage weights for this ci half
        for (int j = tid; j < 16 * TAPS * 16; j += 256) {
            int cih = j & 15; int t = (j >> 4) % TAPS; int col = j / (TAPS * 16);
            float w;
            if (MODE == 0) {
                // Ksum[co0+col][t][half*16+cih]
                w = Wt[(size_t)(co0 + col) * (TAPS * 32) + t * 32 + half * 16 + cih];
            } else {
                // gather Kn[32*win[f]+f][32*win[f']+f'][t] on the fly
                int fp  = half * 16 + cih;
                int cin = NFM * winsh[fp] + fp;
                int fo  = co0 + col;
                int co  = NFM * winsh[fo] + fo;
                w = Wt[((size_t)co * CCH + cin) * TAPS + t];
            }
            wlds[(col * TAPS + t) * CIW + cih] = w;
        }
        __syncthreads();
        #pragma unroll
        for (int q = 0; q < 4; ++q) {
            const int cib = q * 4 + 2 * hi;          // K-pair owned by this lane half
            #pragma unroll
            for (int t = 0; t < TAPS; ++t) {
                const int dy = t / 5, dx = t % 5;
                // A fragment: W[M=lo][K=cib..cib+1]   (contiguous -> ds_load_b64)
                v2f a = *(const v2f*)&wlds[(lo * TAPS + t) * CIW + cib];
                // B fragment: X[K=cib..cib+1][N=lo]   (contiguous -> ds_load_b64)
                v2f bx = *(const v2f*)&patch[((wv + dy) * 20 + (lo + dx)) * CIP
                                             + half * 16 + cib];
                acc = __builtin_amdgcn_wmma_f32_16x16x4_f32(
                        false, a, false, bx, (short)0, acc, false, false);
            }
        }
    }

    if (MODE == 0) {
        // fused reduction of L1*S = (acc/32)*S over the 16 pixels of this tile
        #pragma unroll
        for (int v = 0; v < 8; ++v) {
            int col = v + 8 * hi;
            float val = acc[v] * (S[co0 + col] * (1.0f / NFM));
            float ss = val, mn = val, mx = val;
            for (int m = 1; m < 16; m <<= 1) {
                ss += __shfl_xor(ss, m, 32);
                mn  = fminf(mn, __shfl_xor(mn, m, 32));
                mx  = fmaxf(mx, __shfl_xor(mx, m, 32));
            }
            if (lo == 0) { red[0][wv][col] = ss; red[1][wv][col] = mn; red[2][wv][col] = mx; }
        }
        __syncthreads();
        if (tid < 16) {
            float ss = 0.0f, mn = 3.4e38f, mx = -3.4e38f;
            for (int u = 0; u < 8; ++u) {
                ss += red[0][u][tid];
                mn  = fminf(mn, red[1][u][tid]);
                mx  = fmaxf(mx, red[2][u][tid]);
            }
            int idx = b * CCH + co0 + tid;
            atomicAdd(&sumL[idx], ss);
            atomicMin(&minU[idx], f2ord(mn));
            atomicMax(&maxU[idx], f2ord(mx));
        }
    } else {
        const int r = y0 + wv;
        float* ob = out + (size_t)b * NFM * HW;
        #pragma unroll
        for (int v = 0; v < 8; ++v) {
            int f = co0 + v + 8 * hi;
            ob[(f * 64 + r) * 64 + x0 + lo] = acc[v] * (1.0f / NFM);
        }
    }
}

extern "C" void kernel_launch(void* const* d_in, const int* in_sizes, int n_in,
                              void* d_out, int out_size, void* d_ws, size_t ws_size,
                              hipStream_t stream) {
    const float* A = (const float*)d_in[0];
    const float* K = (const float*)d_in[1];
    const float* S = (const float*)d_in[2];
    float* out = (float*)d_out;

    // workspace carve-up (~0.7 MB total)
    char* ws = (char*)d_ws;
    size_t off = 0;
    auto carve = [&](size_t bytes) -> char* {
        char* p = ws + off; off += (bytes + 255) & ~(size_t)255; return p;
    };
    float*    Kn   = (float*)carve((size_t)CCH * CCH * TAPS * 4);   // 409600
    float*    Ksum = (float*)carve((size_t)CCH * TAPS * NFM * 4);   // 204800
    unsigned* minU = (unsigned*)carve((size_t)BATCH * CCH * 4);
    unsigned* maxU = (unsigned*)carve((size_t)BATCH * CCH * 4);
    float*    sumL = (float*)carve((size_t)BATCH * CCH * 4);
    float*    sumA = (float*)carve((size_t)BATCH * NFM * 4);
    int*      win  = (int*)carve((size_t)BATCH * NFM * 4);

    k_init <<<(BATCH * CCH + 255) / 256, 256, 0, stream>>>(minU, maxU, sumL);
    k_normK<<<(CCH * CCH + 255) / 256, 256, 0, stream>>>(K, Kn);
    k_ksum <<<(CCH * TAPS * NFM + 255) / 256, 256, 0, stream>>>(Kn, Ksum);
    k_sumA <<<BATCH * NFM, 256, 0, stream>>>(A, sumA);

    dim3 g1(32, 4, BATCH);   // 8 row-blocks * 4 col-strips, 4 co-tiles (64 ch)
    k_conv<0><<<g1, 256, 0, stream>>>(A, Ksum, S, nullptr, nullptr, minU, maxU, sumL);

    k_win<<<(BATCH * NFM + 255) / 256, 256, 0, stream>>>(minU, maxU, sumL, sumA, win);

    dim3 g2(32, 2, BATCH);   // 2 co-tiles (32 winning ch)
    k_conv<1><<<g2, 256, 0, stream>>>(A, Kn, nullptr, win, out, nullptr, nullptr, nullptr);
}